// PMEModelWrapper_26946624815172
// MI455X (gfx1250) — compile-verified
//
#include <hip/hip_runtime.h>
#include <math.h>

// ---------------- problem constants ----------------
#define KDIM 128
#define KD2  (KDIM*KDIM)        // 16384
#define KD3  (KDIM*KDIM*KDIM)   // 2097152
#define NLINES KD2
#define NGROUPS (NLINES/16)     // 1024 tile-groups per pass (16 lines each)
#define PI_F 3.14159265358979323846f
#define ALPHA_F 0.35f
#define KE_F 14.3996f

typedef float v2f __attribute__((ext_vector_type(2)));
typedef float v8f __attribute__((ext_vector_type(8)));

// f32 16x16x4 WMMA: D = A(16x4) * B(4x16) + C.  Chain 4 of these for K=16.
__device__ __forceinline__ v8f wmma4(v2f a, v2f b, v8f c) {
  return __builtin_amdgcn_wmma_f32_16x16x4_f32(false, a, false, b, (short)0, c,
                                               false, false);
}

// ---------------- B-spline helpers (P=4) ----------------
__device__ __forceinline__ float bsp2(float t) {
  return fminf(fmaxf(1.f - fabsf(t - 1.f), 0.f), 1.f);
}
__device__ __forceinline__ float bsp3(float t) {
  return 0.5f * (t * bsp2(t) + (3.f - t) * bsp2(t - 1.f));
}
__device__ __forceinline__ float bsp4(float t) {
  return (1.f / 3.f) * (t * bsp3(t) + (4.f - t) * bsp3(t - 1.f));
}
__device__ __forceinline__ float dbsp4(float t) {
  return bsp3(t) - bsp3(t - 1.f);
}

__device__ __forceinline__ void inv3x3(const float* c, float inv[9], float* adet) {
  float a = c[0], b = c[1], cc = c[2];
  float d = c[3], e = c[4], f = c[5];
  float g = c[6], h = c[7], i = c[8];
  float A = e * i - f * h, B = -(d * i - f * g), C = d * h - e * g;
  float det = a * A + b * B + cc * C;
  float id = 1.f / det;
  inv[0] = A * id;              inv[1] = -(b * i - cc * h) * id; inv[2] = (b * f - cc * e) * id;
  inv[3] = B * id;              inv[4] = (a * i - cc * g) * id;  inv[5] = -(a * f - cc * d) * id;
  inv[6] = C * id;              inv[7] = -(a * h - b * g) * id;  inv[8] = (a * e - b * d) * id;
  *adet = fabsf(det);
}

// |b(m)|^-2 Euler spline factor for one dim (P=4): M4(1)=1/6, M4(2)=2/3, M4(3)=1/6
__device__ __forceinline__ float bmod2f(int mi) {
  float m = (float)(mi < KDIM / 2 ? mi : mi - KDIM);
  float ang = 2.f * PI_F * m / (float)KDIM;
  float s1, c1, s2, c2;
  __sincosf(ang, &s1, &c1);
  __sincosf(2.f * ang, &s2, &c2);
  const float M1 = 1.f / 6.f, M2v = 2.f / 3.f, M3v = 1.f / 6.f;
  float Dr = M1 + M2v * c1 + M3v * c2;
  float Di = M2v * s1 + M3v * s2;
  return 1.f / fmaxf(Dr * Dr + Di * Di, 1e-12f);
}

// ---------------- init ----------------
__global__ __launch_bounds__(256) void zeroK(float* __restrict__ Qr,
                                             float* __restrict__ acc) {
  int i = blockIdx.x * blockDim.x + threadIdx.x;
  if (i < KD3) Qr[i] = 0.f;
  if (i < 2) acc[i] = 0.f;
}

// ---------------- charge spreading (scatter) ----------------
__global__ __launch_bounds__(128) void spreadK(const float* __restrict__ pos,
                                               const float* __restrict__ chg,
                                               const float* __restrict__ cell,
                                               float* __restrict__ Qr, int n) {
  int a = blockIdx.x * blockDim.x + threadIdx.x;
  if (a >= n) return;
  float inv[9], det;
  inv3x3(cell, inv, &det);
  float p0 = pos[3 * a], p1 = pos[3 * a + 1], p2 = pos[3 * a + 2];
  float w[3][4];
  int id[3][4];
#pragma unroll
  for (int d = 0; d < 3; ++d) {
    float f = p0 * inv[0 + d] + p1 * inv[3 + d] + p2 * inv[6 + d];
    f -= floorf(f);
    float u = f * (float)KDIM;
    float b = floorf(u);
    int bi = (int)b;
    float du = u - b;
#pragma unroll
    for (int j = 0; j < 4; ++j) {
      w[d][j] = bsp4(du + (float)(3 - j));
      id[d][j] = (bi - 3 + j) & (KDIM - 1);
    }
  }
  float qa = chg[a];
#pragma unroll
  for (int i = 0; i < 4; ++i) {
    int ox = id[0][i] * KD2;
    float wx = qa * w[0][i];
#pragma unroll
    for (int j = 0; j < 4; ++j) {
      int oy = ox + id[1][j] * KDIM;
      float wxy = wx * w[1][j];
#pragma unroll
      for (int kz = 0; kz < 4; ++kz) {
        atomicAdd(&Qr[oy + id[2][kz]], wxy * w[2][kz]);
      }
    }
  }
}

// ---------------- tensorized 128-pt FFT pass (tcFFT: 128 = 16 x 8) ----------------
// One wave handles 2 lines per tile. Stage 1: Y = F16 * X (complex matmul, WMMA).
// Twiddle W128^{k1*n2}, then Stage 2: Z = Y' * blockdiag(F8,F8) (second WMMA matmul).
// - DFT/twiddle factors are tile-invariant -> hoisted to registers in a preamble;
//   the kernel grid-strides over tile groups so the preamble amortizes.
// - Per-wave LDS slice is SoA (real plane / imag plane) so each WMMA fragment is
//   a direct 2-dword DS load with no repacking movs.
// - Geometry (STRIDE,P,Q) is compile-time: div/mod become shifts, and the
//   stride-1 passes get vectorized b128 global stores.
// - Waves touch only their own LDS slice; intra-wave DS ordering makes barriers
//   unnecessary, and the transpose tile aliases the input buffer (program-order WAR).
template <bool REAL, int STRIDE, int P, int Q>
__global__ __launch_bounds__(256) void fft128_pass(const float* __restrict__ srcR,
                                                   const float2* __restrict__ srcC,
                                                   float2* __restrict__ dst,
                                                   float dir) {
  __shared__ float lds[8][512];  // per wave: [0..255] real, [256..511] imag
  const int tid = threadIdx.x;
  const int wave = tid >> 5;
  const int lane = tid & 31;
  float* const re = lds[wave];
  float* const im = lds[wave] + 256;

  const int m = lane & 15;        // A-matrix row (ISA A-layout: M = lane%16)
  const int half = lane >> 4;     // selects which K pair this half-wave holds
  const int n = lane & 15;        // B/C/D column (row-striped across lanes)
  const int n2 = n & 7;
  const int colLine = n >> 3;

  // ---- preamble: all tile-invariant factors into registers (24 sincos total)
  v2f a1R[4], a1I[4];   // stage-1 A fragment: F16[m][kk]
  v2f b2R[4], b2I[4];   // stage-2 B fragment: blockdiag(F8,F8)[kk][n]
  float twc[8], tws[8]; // W128^{k1*n2} per output row
#pragma unroll
  for (int kc = 0; kc < 4; ++kc) {
    int kk0 = kc * 4 + half * 2;
    float s, c;
    __sincosf((2.f * PI_F / 16.f) * (float)((m * kk0) & 15), &s, &c);
    a1R[kc].x = c; a1I[kc].x = dir * s;
    __sincosf((2.f * PI_F / 16.f) * (float)((m * (kk0 + 1)) & 15), &s, &c);
    a1R[kc].y = c; a1I[kc].y = dir * s;
    bool blk0 = ((kk0 >> 3) == (n >> 3));
    bool blk1 = (((kk0 + 1) >> 3) == (n >> 3));
    __sincosf((2.f * PI_F / 8.f) * (float)(((kk0 & 7) * n2) & 7), &s, &c);
    b2R[kc].x = blk0 ? c : 0.f; b2I[kc].x = blk0 ? dir * s : 0.f;
    __sincosf((2.f * PI_F / 8.f) * (float)((((kk0 + 1) & 7) * n2) & 7), &s, &c);
    b2R[kc].y = blk1 ? c : 0.f; b2I[kc].y = blk1 ? dir * s : 0.f;
  }
#pragma unroll
  for (int r = 0; r < 8; ++r) {
    int k1 = r + half * 8;
    float s, c;
    __sincosf((2.f * PI_F / 128.f) * (float)(k1 * n2), &s, &c);
    twc[r] = c; tws[r] = dir * s;
  }

  // ---- grid-stride over tile groups (16 lines per group, 2 lines per wave)
  for (int grp = blockIdx.x; grp < NGROUPS; grp += gridDim.x) {
    const int line0 = grp * 16 + wave * 2;

    // stage lines into this wave's LDS slice (SoA)
#pragma unroll
    for (int j = 0; j < 8; ++j) {
      int i = lane + 32 * j;
      int line = line0 + (i >> 7);
      int base = (line / Q) * P + (line % Q);
      float2 v;
      if (REAL) { v.x = srcR[base + (i & 127) * STRIDE]; v.y = 0.f; }
      else      { v = srcC[base + (i & 127) * STRIDE]; }
      re[i] = v.x;
      im[i] = v.y;
    }

    // stage 1: Y[k1][col] = sum_n1 F16[k1][n1] * x[8*n1+n2]
    v8f aRR = {}, aII = {}, aCI = {};
    const int cbase = colLine * 128 + n2;
#pragma unroll
    for (int kc = 0; kc < 4; ++kc) {
      int o = cbase + (kc * 4 + half * 2) * 8;
      v2f bR; bR.x = re[o]; bR.y = re[o + 8];
      v2f bI; bI.x = im[o]; bI.y = im[o + 8];
      aRR = wmma4(a1R[kc], bR, aRR);   // Re*Re
      aII = wmma4(a1I[kc], bI, aII);   // Im*Im
      aCI = wmma4(a1R[kc], bI, aCI);   // Re*Im
      aCI = wmma4(a1I[kc], bR, aCI);   // + Im*Re
    }

    // twiddle + transpose through LDS (C/D row M = r + half*8); aliases input buf
#pragma unroll
    for (int r = 0; r < 8; ++r) {
      int k1 = r + half * 8;
      float yr = aRR[r] - aII[r];
      float yi = aCI[r];
      re[k1 * 16 + n] = yr * twc[r] - yi * tws[r];
      im[k1 * 16 + n] = yr * tws[r] + yi * twc[r];
    }

    // stage 2: Z = Y' * blockdiag(F8, F8)
    v8f dRR = {}, dII = {}, dCI = {};
#pragma unroll
    for (int kc = 0; kc < 4; ++kc) {
      int o = m * 16 + kc * 4 + half * 2;
      v2f aR; aR.x = re[o]; aR.y = re[o + 1];
      v2f aI; aI.x = im[o]; aI.y = im[o + 1];
      dRR = wmma4(aR, b2R[kc], dRR);
      dII = wmma4(aI, b2I[kc], dII);
      dCI = wmma4(aR, b2I[kc], dCI);
      dCI = wmma4(aI, b2R[kc], dCI);
    }

    // write out: X[k1 + 16*k2] of line (colLine)
    const int lineO = line0 + colLine;
    const int baseO = (lineO / Q) * P + (lineO % Q);
#pragma unroll
    for (int r = 0; r < 8; ++r) {
      int k1 = r + half * 8;
      float2 z;
      z.x = dRR[r] - dII[r];
      z.y = dCI[r];
      dst[baseO + (k1 + 16 * n2) * STRIDE] = z;
    }
  }
}

// ---------------- spectral scale + energy reduction ----------------
__global__ __launch_bounds__(256) void recipScaleK(float2* __restrict__ Cg,
                                                   const float* __restrict__ cell,
                                                   float* __restrict__ acc) {
  __shared__ float red[256];
  int idx = blockIdx.x * 256 + threadIdx.x;
  float inv[9], det;
  inv3x3(cell, inv, &det);
  int ix = idx >> 14, iy = (idx >> 7) & 127, iz = idx & 127;
  float m0 = (float)(ix < 64 ? ix : ix - 128);
  float m1 = (float)(iy < 64 ? iy : iy - 128);
  float m2 = (float)(iz < 64 ? iz : iz - 128);
  float kx = 2.f * PI_F * (m0 * inv[0] + m1 * inv[1] + m2 * inv[2]);
  float ky = 2.f * PI_F * (m0 * inv[3] + m1 * inv[4] + m2 * inv[5]);
  float kz = 2.f * PI_F * (m0 * inv[6] + m1 * inv[7] + m2 * inv[8]);
  float k2 = kx * kx + ky * ky + kz * kz;
  float G = 0.f;
  if (k2 > 0.f) G = __expf(-k2 / (4.f * ALPHA_F * ALPHA_F)) / k2;
  float b3 = bmod2f(ix) * bmod2f(iy) * bmod2f(iz);
  float C0 = KE_F * 2.f * PI_F / det;
  float2 S = Cg[idx];
  float e = C0 * G * b3 * (S.x * S.x + S.y * S.y);
  float sc = 2.f * C0 * G * b3;   // dE/dQ via unnormalized inverse FFT
  S.x *= sc; S.y *= sc;
  Cg[idx] = S;
  red[threadIdx.x] = e;
  __syncthreads();
  for (int s = 128; s > 0; s >>= 1) {
    if (threadIdx.x < s) red[threadIdx.x] += red[threadIdx.x + s];
    __syncthreads();
  }
  if (threadIdx.x == 0) atomicAdd(acc, red[0]);
}

// ---------------- self-energy (sum q^2) ----------------
__global__ __launch_bounds__(256) void selfK(const float* __restrict__ chg, int n,
                                             float* __restrict__ acc) {
  __shared__ float red[256];
  float s = 0.f;
  for (int i = blockIdx.x * 256 + threadIdx.x; i < n; i += gridDim.x * 256) {
    float q = chg[i];
    s += q * q;
  }
  red[threadIdx.x] = s;
  __syncthreads();
  for (int st = 128; st > 0; st >>= 1) {
    if (threadIdx.x < st) red[threadIdx.x] += red[threadIdx.x + st];
    __syncthreads();
  }
  if (threadIdx.x == 0) atomicAdd(acc, red[0]);
}

// ---------------- force gather ----------------
__global__ __launch_bounds__(128) void gatherK(const float* __restrict__ pos,
                                               const float* __restrict__ chg,
                                               const float* __restrict__ cell,
                                               const float2* __restrict__ phi,
                                               float* __restrict__ out, int n) {
  int a = blockIdx.x * blockDim.x + threadIdx.x;
  if (a >= n) return;
  float inv[9], det;
  inv3x3(cell, inv, &det);
  float p0 = pos[3 * a], p1 = pos[3 * a + 1], p2 = pos[3 * a + 2];
  float w[3][4], dw[3][4];
  int id[3][4];
#pragma unroll
  for (int d = 0; d < 3; ++d) {
    float f = p0 * inv[0 + d] + p1 * inv[3 + d] + p2 * inv[6 + d];
    f -= floorf(f);
    float u = f * (float)KDIM;
    float b = floorf(u);
    int bi = (int)b;
    float du = u - b;
#pragma unroll
    for (int j = 0; j < 4; ++j) {
      float t = du + (float)(3 - j);
      w[d][j] = bsp4(t);
      dw[d][j] = dbsp4(t);
      id[d][j] = (bi - 3 + j) & (KDIM - 1);
    }
  }
  float g0 = 0.f, g1 = 0.f, g2 = 0.f;
#pragma unroll
  for (int i = 0; i < 4; ++i) {
    int ox = id[0][i] * KD2;
#pragma unroll
    for (int j = 0; j < 4; ++j) {
      int oy = ox + id[1][j] * KDIM;
#pragma unroll
      for (int kz = 0; kz < 4; ++kz) {
        float ph = phi[oy + id[2][kz]].x;   // Re part of inverse transform
        g0 += dw[0][i] * w[1][j] * w[2][kz] * ph;
        g1 += w[0][i] * dw[1][j] * w[2][kz] * ph;
        g2 += w[0][i] * w[1][j] * dw[2][kz] * ph;
      }
    }
  }
  float qa = chg[a];
#pragma unroll
  for (int c = 0; c < 3; ++c) {
    float gp = qa * (float)KDIM *
               (g0 * inv[c * 3 + 0] + g1 * inv[c * 3 + 1] + g2 * inv[c * 3 + 2]);
    out[1 + 3 * a + c] = -gp;   // force = -dE/dr
  }
}

// ---------------- final combine ----------------
__global__ void combineK(const float* __restrict__ acc, float* __restrict__ out) {
  out[0] = acc[0] - (KE_F * ALPHA_F / sqrtf(PI_F)) * acc[1];
}

// ---------------- host launch ----------------
extern "C" void kernel_launch(void* const* d_in, const int* in_sizes, int n_in,
                              void* d_out, int out_size, void* d_ws, size_t ws_size,
                              hipStream_t stream) {
  const float* pos = (const float*)d_in[0];
  const float* chg = (const float*)d_in[1];
  const float* cell = (const float*)d_in[2];
  float* out = (float*)d_out;
  float* ws = (float*)d_ws;
  int n = in_sizes[0] / 3;

  // workspace layout (floats): Qr[KD3] | C1[2*KD3] | C2[2*KD3] | acc[2]  (~42 MB)
  float* Qr = ws;
  float2* C1 = (float2*)(ws + KD3);
  float2* C2 = (float2*)(ws + 3 * (size_t)KD3);
  float* acc = ws + 5 * (size_t)KD3;

  const int FFT_BLOCKS = 256;   // each block loops over 4 tile groups

  zeroK<<<(KD3 + 255) / 256, 256, 0, stream>>>(Qr, acc);
  spreadK<<<(n + 127) / 128, 128, 0, stream>>>(pos, chg, cell, Qr, n);

  // forward FFT: z (stride 1), y (stride 128), x (stride 16384)
  fft128_pass<true, 1, KDIM, 1><<<FFT_BLOCKS, 256, 0, stream>>>(Qr, nullptr, C1, -1.f);
  fft128_pass<false, KDIM, KD2, KDIM><<<FFT_BLOCKS, 256, 0, stream>>>(nullptr, C1, C2, -1.f);
  fft128_pass<false, KD2, 0, KD2><<<FFT_BLOCKS, 256, 0, stream>>>(nullptr, C2, C1, -1.f);

  recipScaleK<<<KD3 / 256, 256, 0, stream>>>(C1, cell, acc);

  // inverse FFT (unnormalized), any axis order
  fft128_pass<false, KD2, 0, KD2><<<FFT_BLOCKS, 256, 0, stream>>>(nullptr, C1, C2, 1.f);
  fft128_pass<false, KDIM, KD2, KDIM><<<FFT_BLOCKS, 256, 0, stream>>>(nullptr, C2, C1, 1.f);
  fft128_pass<false, 1, KDIM, 1><<<FFT_BLOCKS, 256, 0, stream>>>(nullptr, C1, C2, 1.f);

  selfK<<<256, 256, 0, stream>>>(chg, n, acc + 1);
  gatherK<<<(n + 127) / 128, 128, 0, stream>>>(pos, chg, cell, C2, out, n);
  combineK<<<1, 1, 0, stream>>>(acc, out);
}